// MarginRankingLoss_90013924589720
// MI455X (gfx1250) — compile-verified
//
#include <hip/hip_runtime.h>
#include <hip/hip_bf16.h>

// MarginRankingLoss over (pos, neg) doc pairs, B=4096 rows, D=128 docs/row.
//
// Strategy (MI455X / gfx1250, wave32):
//   M[i,j] = a_i + b_j is rank-1 + const  ->  each 16x16 tile is ONE
//   v_wmma_f32_16x16x4_f32 (A col0 = a, col1 = ones; B row0 = ones, row1 = b).
//   Masking is folded linearly via -1e30 sentinels so relu kills masked pairs.
//   Only relu + accumulate remain on VALU (16 ops / 256 pairs); the rank-1
//   expansion runs on the XDL matrix pipe and co-executes with VALU.
//   One wave per row; 4096 waves total; f32 end to end (no precision loss).

typedef __attribute__((ext_vector_type(2))) float v2f;
typedef __attribute__((ext_vector_type(8))) float v8f;

#define MARGIN_F 0.5f
#define NEGBIG  (-1.0e30f)

__global__ __launch_bounds__(256) void
margin_rank_rows_kernel(const float* __restrict__ logits,
                        const int*   __restrict__ labels,
                        float* __restrict__ ws, int rows) {
  const int D = 128;
  const int lane = threadIdx.x & 31;
  const int wave = threadIdx.x >> 5;
  const int row  = blockIdx.x * 8 + wave;
  if (row >= rows) return;  // wave-uniform: EXEC stays all-ones for live waves

  const float* lg = logits + (size_t)row * D;
  const int*   lb = labels + (size_t)row * D;

  // Lane L owns elements L, L+32, L+64, L+96 (coalesced 128B per slot).
  float a[4], b[4];
  int cp = 0;
  #pragma unroll
  for (int s = 0; s < 4; ++s) {
    const int k = lane + 32 * s;
    const float x = lg[k];
    const bool pos = (lb[k] > 0);
    a[s] = pos ? (MARGIN_F - x) : NEGBIG;  // pos-masked (margin - l_i)
    b[s] = pos ? NEGBIG : x;               // neg-masked l_j
    cp += pos ? 1 : 0;
  }
  // npos across the row (fixed-order tree -> deterministic)
  #pragma unroll
  for (int o = 16; o; o >>= 1) cp += __shfl_xor(cp, o, 32);
  const float counts = (float)cp * (float)(128 - cp);  // npos * nneg, exact

  // Build WMMA operands for all 8 sixteen-element chunks.
  // A 16x4 f32 layout (ISA 7.12.2): VGPR0 = K0 (lanes 0-15) | K2 (16-31),
  //                                 VGPR1 = K1 | K3.
  // B 4x16 assumed dual:            VGPR0 = K0 | K2, VGPR1 = K1 | K3.
  const bool lo = lane < 16;
  const int  lsel = lane & 15;
  v2f Aop[8], Bop[8];
  #pragma unroll
  for (int t = 0; t < 8; ++t) {
    const int src = ((t & 1) << 4) | lsel;   // chunk t lives in slot t>>1, half t&1
    const float av = __shfl(a[t >> 1], src, 32);
    const float bv = __shfl(b[t >> 1], src, 32);
    Aop[t][0] = lo ? av   : 0.0f;  // A[:,0] = a   ; A[:,2] = 0
    Aop[t][1] = lo ? 1.0f : 0.0f;  // A[:,1] = 1   ; A[:,3] = 0
    Bop[t][0] = lo ? 1.0f : 0.0f;  // B[0,:] = 1   ; B[2,:] = 0
    Bop[t][1] = lo ? bv   : 0.0f;  // B[1,:] = b   ; B[3,:] = 0
  }

  const v8f zero = {0.f, 0.f, 0.f, 0.f, 0.f, 0.f, 0.f, 0.f};
  v8f acc = zero;
  for (int ti = 0; ti < 8; ++ti) {
    #pragma unroll
    for (int tj = 0; tj < 8; ++tj) {
      // D[i,j] = a_i + b_j for this 16x16 tile, on the XDL pipe.
      v8f d = __builtin_amdgcn_wmma_f32_16x16x4_f32(
          false, Aop[ti], false, Bop[tj], (short)0, zero, false, false);
      #pragma unroll
      for (int q = 0; q < 8; ++q)
        acc[q] += fmaxf(d[q], 0.0f);   // relu + accumulate (VALU, co-exec)
    }
  }

  float s = 0.0f;
  #pragma unroll
  for (int q = 0; q < 8; ++q) s += acc[q];
  #pragma unroll
  for (int o = 16; o; o >>= 1) s += __shfl_xor(s, o, 32);

  if (lane == 0) {
    const bool valid = counts > 0.0f;
    ws[row]        = valid ? (s / counts) : 0.0f;
    ws[rows + row] = valid ? 1.0f : 0.0f;
  }
}

__global__ __launch_bounds__(256) void
margin_rank_final_kernel(const float* __restrict__ ws, float* __restrict__ out,
                         int rows) {
  const int tid = threadIdx.x;
  float sm = 0.0f, sv = 0.0f;
  for (int r = tid; r < rows; r += 256) {  // fixed traversal -> deterministic
    sm += ws[r];
    sv += ws[rows + r];
  }
  #pragma unroll
  for (int o = 16; o; o >>= 1) {
    sm += __shfl_xor(sm, o, 32);
    sv += __shfl_xor(sv, o, 32);
  }
  __shared__ float sh[16];
  const int w = tid >> 5, l = tid & 31;
  if (l == 0) { sh[w] = sm; sh[8 + w] = sv; }
  __syncthreads();
  if (tid == 0) {
    float tm = 0.0f, tv = 0.0f;
    #pragma unroll
    for (int i = 0; i < 8; ++i) { tm += sh[i]; tv += sh[8 + i]; }
    out[0] = (tv > 0.0f) ? (tm / tv) : 0.0f;
  }
}

extern "C" void kernel_launch(void* const* d_in, const int* in_sizes, int n_in,
                              void* d_out, int out_size, void* d_ws, size_t ws_size,
                              hipStream_t stream) {
  const float* logits = (const float*)d_in[0];
  const int*   labels = (const int*)d_in[1];
  float* out = (float*)d_out;
  float* ws  = (float*)d_ws;

  const int D = 128;
  const int rows = in_sizes[0] / D;  // 4096

  const int blocks = (rows + 7) / 8;  // 8 waves (rows) per 256-thread block
  margin_rank_rows_kernel<<<blocks, 256, 0, stream>>>(logits, labels, ws, rows);
  margin_rank_final_kernel<<<1, 256, 0, stream>>>(ws, out, rows);
}